// PyramidROIAlign_54674933678551
// MI455X (gfx1250) — compile-verified
//
#include <hip/hip_runtime.h>
#include <stdint.h>

// PyramidROIAlign for MI455X (gfx1250).
// Memory-bound gather op (~0.4 FLOP/byte): no matrix structure -> no WMMA.
// CDNA5 paths used:
//  - global_load_async_to_lds_b128 + s_wait_asynccnt: double-buffered corner
//    gathers through the async engine (no VGPR destinations, latency overlap).
//  - non-temporal output stores: the 178 MB pyramid stays resident in the
//    192 MB L2 while the write-once 100 MB output streams past it.

#define POOL 7
#define CCH  256
#define C4   (CCH / 4)   // 64 float4 lanes per position

#if __has_builtin(__builtin_amdgcn_global_load_async_to_lds_b128) && \
    __has_builtin(__builtin_amdgcn_s_wait_asynccnt)
#define USE_ASYNC_LDS 1
#else
#define USE_ASYNC_LDS 0
#endif

typedef int   v4i __attribute__((vector_size(16)));  // builtin param type: int __vector(4)
typedef float v4f __attribute__((vector_size(16)));

// Inner (void*) drops const; outer cast changes pointee type and performs the
// generic->addrspace(1)/(3) addrspacecast.
#define GPTR(x) ((__attribute__((address_space(1))) v4i*)(void*)(x))
#define LPTR(x) ((__attribute__((address_space(3))) v4i*)(void*)(x))

__device__ __forceinline__ v4f lerp4(v4f a, v4f b, float w) {
    return a + w * (b - a);   // matches reference: v0 + w*(v1-v0), elementwise
}

__global__ __launch_bounds__(256)
void PyramidROIAlign_54674933678551_kernel(
    const float* __restrict__ boxes,
    const float* __restrict__ p2, const float* __restrict__ p3,
    const float* __restrict__ p4, const float* __restrict__ p5,
    float* __restrict__ out, int nper /* boxes per batch */)
{
    const int roi = blockIdx.x;
    const int b   = roi / nper;

    // ---- box + level selection (wave-uniform -> lowered to scalar-float SALU) ----
    const float* bp  = boxes + (size_t)roi * 4;
    const float  by1 = bp[0], bx1 = bp[1], by2 = bp[2], bx2 = bp[3];
    const float  bh  = by2 - by1, bw = bx2 - bx1;

    // roi_level = clip(round(log2(sqrt(max(h*w,1e-12)) / (224/1024))), 2, 5)
    const float s   = sqrtf(fmaxf(bh * bw, 1e-12f));
    const float lf  = rintf(log2f(s / 0.21875f));  // rintf == round-half-even == jnp.round
    const int   lvl = (int)fminf(fmaxf(lf, 2.0f), 5.0f);

    const float* fp; int H;
    if      (lvl <= 2) { fp = p2; H = 256; }
    else if (lvl == 3) { fp = p3; H = 128; }
    else if (lvl == 4) { fp = p4; H = 64;  }
    else               { fp = p5; H = 32;  }
    fp += (size_t)b * (size_t)H * (size_t)H * CCH;

    // ---- per-ROI sample tables in LDS ----
    __shared__ int   sy0[POOL], sy1[POOL], sx0[POOL], sx1[POOL];
    __shared__ float swy[POOL], swx[POOL];

    if (threadIdx.x < POOL) {
        const int   i   = threadIdx.x;
        const float t   = (float)i / 6.0f;             // linspace(0,1,7)
        const float hm1 = (float)(H - 1);

        const float yv  = (by1 + bh * t) * hm1;
        const float y0f = floorf(yv);
        swy[i] = yv - y0f;
        const int yi = (int)y0f;
        sy0[i] = min(H - 1, max(0, yi));
        sy1[i] = min(H - 1, max(0, yi + 1));

        const float xv  = (bx1 + bw * t) * hm1;
        const float x0f = floorf(xv);
        swx[i] = xv - x0f;
        const int xi = (int)x0f;
        sx0[i] = min(H - 1, max(0, xi));
        sx1[i] = min(H - 1, max(0, xi + 1));
    }
    __syncthreads();

    const int tid = threadIdx.x;
    const int c4  = tid & (C4 - 1);   // channel float4 lane 0..63
    const int grp = tid >> 6;         // 0..3 (each group = 2 full wave32s)

    const v4f* __restrict__ fb = (const v4f*)fp;
    v4f* __restrict__ ob = (v4f*)(out + (size_t)roi * (POOL * POOL * CCH));

#if USE_ASYNC_LDS
    // Double-buffered async gather: [buf 0/1][corner 0..3][tid] float4 slots.
    // Each lane reads back only slots it filled -> per-wave s_wait_asynccnt
    // is the only synchronization required (no barrier in the pipeline).
    __shared__ v4f sbuf[8 * 256];

    auto issue = [&](int p, int buf) {
        const int py = p / POOL, px = p - py * POOL;
        const int r0 = sy0[py] * H, r1 = sy1[py] * H;
        const int i00 = (r0 + sx0[px]) * C4 + c4;
        const int i01 = (r0 + sx1[px]) * C4 + c4;
        const int i10 = (r1 + sx0[px]) * C4 + c4;
        const int i11 = (r1 + sx1[px]) * C4 + c4;
        const int sb  = buf * 4 * 256 + tid;
        __builtin_amdgcn_global_load_async_to_lds_b128(
            GPTR(fb + i00), LPTR(&sbuf[sb + 0 * 256]), 0, 0);
        __builtin_amdgcn_global_load_async_to_lds_b128(
            GPTR(fb + i01), LPTR(&sbuf[sb + 1 * 256]), 0, 0);
        __builtin_amdgcn_global_load_async_to_lds_b128(
            GPTR(fb + i10), LPTR(&sbuf[sb + 2 * 256]), 0, 0);
        __builtin_amdgcn_global_load_async_to_lds_b128(
            GPTR(fb + i11), LPTR(&sbuf[sb + 3 * 256]), 0, 0);
    };

    int buf = 0;
    issue(grp, 0);                               // prologue (grp < 4 <= 48)
    for (int p = grp; p < POOL * POOL; p += 4) {
        const int pn = p + 4;
        if (pn < POOL * POOL) {                  // wave-uniform branch
            issue(pn, buf ^ 1);
            __builtin_amdgcn_s_wait_asynccnt(4); // current buffer's 4 loads done
        } else {
            __builtin_amdgcn_s_wait_asynccnt(0);
        }
        asm volatile("" ::: "memory");           // keep LDS reads below the wait

        const int sb = buf * 4 * 256 + tid;
        const v4f v00 = sbuf[sb + 0 * 256];
        const v4f v01 = sbuf[sb + 1 * 256];
        const v4f v10 = sbuf[sb + 2 * 256];
        const v4f v11 = sbuf[sb + 3 * 256];

        const int py = p / POOL, px = p - py * POOL;
        const float fx = swx[px], fy = swy[py];
        const v4f top = lerp4(v00, v01, fx);
        const v4f bot = lerp4(v10, v11, fx);
        __builtin_nontemporal_store(lerp4(top, bot, fy), &ob[p * C4 + c4]);
        buf ^= 1;
    }
#else
    // Fallback: direct coalesced float4 gathers + gfx1250 prefetch of the
    // next position's corner line.
    for (int p = grp; p < POOL * POOL; p += 4) {
        const int py = p / POOL, px = p - py * POOL;
        const int r0 = sy0[py] * H, r1 = sy1[py] * H;
        const v4f v00 = fb[(r0 + sx0[px]) * C4 + c4];
        const v4f v01 = fb[(r0 + sx1[px]) * C4 + c4];
        const v4f v10 = fb[(r1 + sx0[px]) * C4 + c4];
        const v4f v11 = fb[(r1 + sx1[px]) * C4 + c4];

        const int pn = p + 4;
        if (pn < POOL * POOL) {
            const int py2 = pn / POOL, px2 = pn - py2 * POOL;
            __builtin_prefetch((const void*)(fb + (sy0[py2] * H + sx0[px2]) * C4 + c4), 0, 1);
            __builtin_prefetch((const void*)(fb + (sy1[py2] * H + sx0[px2]) * C4 + c4), 0, 1);
        }

        const float fx = swx[px], fy = swy[py];
        const v4f top = lerp4(v00, v01, fx);
        const v4f bot = lerp4(v10, v11, fx);
        __builtin_nontemporal_store(lerp4(top, bot, fy), &ob[p * C4 + c4]);
    }
#endif
}

extern "C" void kernel_launch(void* const* d_in, const int* in_sizes, int n_in,
                              void* d_out, int out_size, void* d_ws, size_t ws_size,
                              hipStream_t stream) {
    const float* boxes = (const float*)d_in[0];
    const float* p2    = (const float*)d_in[1];
    const float* p3    = (const float*)d_in[2];
    const float* p4    = (const float*)d_in[3];
    const float* p5    = (const float*)d_in[4];
    float* out = (float*)d_out;

    const int nbox_total = in_sizes[0] / 4;                       // B*N
    const int B = (int)((long long)in_sizes[1] / (256LL * 256 * 256));
    const int N = nbox_total / (B > 0 ? B : 1);

    dim3 grid(nbox_total), block(256);
    hipLaunchKernelGGL(PyramidROIAlign_54674933678551_kernel,
                       grid, block, 0, stream,
                       boxes, p2, p3, p4, p5, out, N);
}